// SystemDE_84421877170856
// MI455X (gfx1250) — compile-verified
//
#include <hip/hip_runtime.h>
#include <math.h>

typedef float v2f __attribute__((ext_vector_type(2)));
typedef float v8f __attribute__((ext_vector_type(8)));

#define Bc 64
#define Tc 2048
#define Nc 128
#define Cc 32
#define Mc 128

// Hardware transcendentals (TRANS ops co-execute with XDL/WMMA on CDNA5).
__device__ __forceinline__ float fast_tanh(float x) {
#if __has_builtin(__builtin_amdgcn_tanhf)
    return __builtin_amdgcn_tanhf(x);          // v_tanh_f32
#else
    return tanhf(x);
#endif
}

__device__ __forceinline__ float softplus_f(float x) {
#if __has_builtin(__builtin_amdgcn_exp2f) && __has_builtin(__builtin_amdgcn_logf)
    // log1p(exp(x)) via hardware base-2 exp/log
    const float e = __builtin_amdgcn_exp2f(x * 1.44269504088896340736f);  // v_exp_f32
    const float r = 0.69314718055994530942f * __builtin_amdgcn_logf(1.0f + e); // v_log_f32
    return (x > 20.0f) ? x : r;
#else
    return (x > 20.0f) ? x : log1pf(expf(x));
#endif
}

// Fused SDE scan + output projection.
// Grid: B/16 = 4 blocks. Block: 256 threads = 8 waves.
// Wave w computes the (16 batch rows) x (16 cols [16w..16w+15]) tile of both
// the drift matmul (cols of N) and the obs matmul (cols of M).
__global__ __launch_bounds__(256, 1) void sde_fused_kernel(
    const float* __restrict__ ts, const float* __restrict__ control,
    const float* __restrict__ dWn, const float* __restrict__ Wh,
    const float* __restrict__ Wu, const float* __restrict__ bh,
    const float* __restrict__ sigma, const float* __restrict__ Wo,
    const float* __restrict__ bo, float* __restrict__ out)
{
    const int tid  = threadIdx.x;
    const int lane = tid & 31;
    const int wv   = tid >> 5;     // wave id 0..7
    const int n0   = wv * 16;      // this wave's output-column tile base
    const int lr   = lane & 15;    // 0..15
    const int lh   = lane >> 4;    // 0 or 1
    const int b0   = blockIdx.x * 16;

    // y double buffer: substep1 reads [0] writes [1]; substep2 reads [1] writes [0].
    __shared__ float ybuf[2][16][132];   // padded rows: conflict-free ds_load_b64
    __shared__ float xbuf[2][16][40];    // u = [t, control(32)], cols 33..35 stay 0

    for (int e = tid; e < 2 * 16 * 132; e += 256) ((float*)ybuf)[e] = 0.0f;
    for (int e = tid; e < 2 * 16 * 40;  e += 256) ((float*)xbuf)[e] = 0.0f;

    const float dt  = ts[1] - ts[0];
    const float dt0 = 0.5f * dt;
    const float gs  = sigma[0] * sqrtf(dt0);

    // ---- Persistent B-matrix fragments held in VGPRs for the whole scan ----
    // B-frag layout for V_WMMA_F32_16X16X4_F32 (4x16 tile, 2 VGPRs):
    //   vgpr0: lanes0-15 -> K=0, lanes16-31 -> K=2 ; vgpr1: K=1 / K=3
    // Per lane (col = n0+lr): frag[kk] = { W[col][4kk+2*lh], W[col][4kk+2*lh+1] }
    v2f fWh[32], fWo[32], fWu[9];
    {
        const int row = n0 + lr;
        #pragma unroll
        for (int kk = 0; kk < 32; ++kk) {
            const int c = 4 * kk + 2 * lh;
            fWh[kk].x = Wh[row * Nc + c];
            fWh[kk].y = Wh[row * Nc + c + 1];
            fWo[kk].x = Wo[row * Nc + c];
            fWo[kk].y = Wo[row * Nc + c + 1];
        }
        #pragma unroll
        for (int kk = 0; kk < 9; ++kk) {
            const int c = 4 * kk + 2 * lh;
            fWu[kk].x = (c     < Cc + 1) ? Wu[row * (Cc + 1) + c]     : 0.0f;
            fWu[kk].y = (c + 1 < Cc + 1) ? Wu[row * (Cc + 1) + c + 1] : 0.0f;
        }
    }
    const float bhv = bh[n0 + lr];
    const float bov = bo[n0 + lr];

    // x at k = 0
    for (int e = tid; e < 16 * Cc; e += 256) {
        const int r = e >> 5, c = e & 31;
        xbuf[0][r][1 + c] = control[((size_t)(b0 + r) * Tc + 0) * Cc + c];
    }
    if (tid < 16) xbuf[0][tid][0] = ts[0];
    __syncthreads();

    // Wave-private y tile (same element<->lane mapping as the WMMA C/D tile).
    v8f yreg = {0.f, 0.f, 0.f, 0.f, 0.f, 0.f, 0.f, 0.f};

    const size_t S = (size_t)Bc * Tc * Mc;

    // Per-lane base pointers; per-row constants folded at compile time.
    const size_t RSdw = (size_t)(Tc - 1) * 2 * Nc;   // dW batch-row stride (elems)
    const float* dwp  = dWn + (size_t)(b0 + 8 * lh) * RSdw + n0 + lr;
    float*       outp = out + (size_t)(b0 + 8 * lh) * Tc * Mc + n0 + lr;

    // Outputs at t = 0 (y0 == 0 -> obs = bo)
    {
        const float o  = bov;
        const float rt = softplus_f(o);
        #pragma unroll
        for (int v = 0; v < 8; ++v) {
            const size_t idx = (size_t)v * ((size_t)Tc * Mc);
            outp[idx] = rt; outp[S + idx] = rt; outp[2 * S + idx] = o;
        }
    }

    int cx = 0;
    size_t kdw = 0;                         // element offset of dW[., k, 0, .]
    #pragma unroll 1
    for (int k = 0; k < Tc - 1; ++k) {
        const int nx = cx ^ 1;

        // Stage x[k+1] into the other x buffer (consumed after barrier 1).
        for (int e = tid; e < 16 * Cc; e += 256) {
            const int r = e >> 5, c = e & 31;
            xbuf[nx][r][1 + c] = control[((size_t)(b0 + r) * Tc + (k + 1)) * Cc + c];
        }
        if (tid < 16) xbuf[nx][tid][0] = ts[k + 1];

        // Hoist both substeps' noise loads: the 41-WMMA drift chain below
        // covers their HBM latency.
        float dw0[8], dw1[8];
        #pragma unroll
        for (int v = 0; v < 8; ++v) {
            const size_t o8 = (size_t)v * RSdw + kdw;
            dw0[v] = dwp[o8];
            dw1[v] = dwp[o8 + Nc];
        }
        // Keep the dW stream ahead in L2 (global_prefetch_b8).
        if (k + 1 < Tc - 1) {
            #pragma unroll
            for (int v = 0; v < 8; ++v)
                __builtin_prefetch(dwp + (size_t)v * RSdw + kdw + 2 * Nc, 0, 0);
        }

        // ---------------- substep 1: u = x[k], reads ybuf[0] ----------------
        {
            v8f acc;
            #pragma unroll
            for (int v = 0; v < 8; ++v) acc[v] = bhv;
            #pragma unroll
            for (int kk = 0; kk < 32; ++kk) {
                v2f a = *(const v2f*)&ybuf[0][lr][4 * kk + 2 * lh];
                acc = __builtin_amdgcn_wmma_f32_16x16x4_f32(
                    false, a, false, fWh[kk], (short)0, acc, false, false);
            }
            #pragma unroll
            for (int kk = 0; kk < 9; ++kk) {
                v2f a = *(const v2f*)&xbuf[cx][lr][4 * kk + 2 * lh];
                acc = __builtin_amdgcn_wmma_f32_16x16x4_f32(
                    false, a, false, fWu[kk], (short)0, acc, false, false);
            }
            #pragma unroll
            for (int v = 0; v < 8; ++v)
                yreg[v] = yreg[v] + fast_tanh(acc[v]) * dt0 + gs * dw0[v];
            #pragma unroll
            for (int v = 0; v < 8; ++v) ybuf[1][v + 8 * lh][n0 + lr] = yreg[v];
            __syncthreads();  // barrier 1: publish mid-state
        }

        // -------- substep 2: u = 0.5*(x[k]+x[k+1]), reads ybuf[1] -----------
        {
            v8f acc;
            #pragma unroll
            for (int v = 0; v < 8; ++v) acc[v] = bhv;
            #pragma unroll
            for (int kk = 0; kk < 32; ++kk) {
                v2f a = *(const v2f*)&ybuf[1][lr][4 * kk + 2 * lh];
                acc = __builtin_amdgcn_wmma_f32_16x16x4_f32(
                    false, a, false, fWh[kk], (short)0, acc, false, false);
            }
            #pragma unroll
            for (int kk = 0; kk < 9; ++kk) {
                v2f a0 = *(const v2f*)&xbuf[cx][lr][4 * kk + 2 * lh];
                v2f a1 = *(const v2f*)&xbuf[nx][lr][4 * kk + 2 * lh];
                v2f a;
                a.x = 0.5f * (a0.x + a1.x);
                a.y = 0.5f * (a0.y + a1.y);
                acc = __builtin_amdgcn_wmma_f32_16x16x4_f32(
                    false, a, false, fWu[kk], (short)0, acc, false, false);
            }
            #pragma unroll
            for (int v = 0; v < 8; ++v)
                yreg[v] = yreg[v] + fast_tanh(acc[v]) * dt0 + gs * dw1[v];
            #pragma unroll
            for (int v = 0; v < 8; ++v) ybuf[0][v + 8 * lh][n0 + lr] = yreg[v];
            __syncthreads();  // barrier 2: publish y[k+1]
        }

        // -------- fused output projection for t = k+1 (off critical path) ---
        {
            v8f acc;
            #pragma unroll
            for (int v = 0; v < 8; ++v) acc[v] = bov;
            #pragma unroll
            for (int kk = 0; kk < 32; ++kk) {
                v2f a = *(const v2f*)&ybuf[0][lr][4 * kk + 2 * lh];
                acc = __builtin_amdgcn_wmma_f32_16x16x4_f32(
                    false, a, false, fWo[kk], (short)0, acc, false, false);
            }
            const size_t trow = (size_t)(k + 1) * Mc;
            #pragma unroll
            for (int v = 0; v < 8; ++v) {
                const float o  = acc[v];
                const float rt = softplus_f(o);
                const size_t idx = (size_t)v * ((size_t)Tc * Mc) + trow;
                outp[idx] = rt; outp[S + idx] = rt; outp[2 * S + idx] = o;
            }
        }
        cx = nx;
        kdw += 2 * Nc;
    }
}

extern "C" void kernel_launch(void* const* d_in, const int* in_sizes, int n_in,
                              void* d_out, int out_size, void* d_ws, size_t ws_size,
                              hipStream_t stream) {
    (void)in_sizes; (void)n_in; (void)out_size; (void)d_ws; (void)ws_size;
    const float* ts      = (const float*)d_in[0];
    const float* control = (const float*)d_in[1];
    const float* dWn     = (const float*)d_in[2];
    const float* Wh      = (const float*)d_in[3];
    const float* Wu      = (const float*)d_in[4];
    const float* bh      = (const float*)d_in[5];
    const float* sigma   = (const float*)d_in[6];
    const float* Wo      = (const float*)d_in[7];
    const float* bo      = (const float*)d_in[8];
    float* out = (float*)d_out;

    dim3 grid(Bc / 16), block(256);
    hipLaunchKernelGGL(sde_fused_kernel, grid, block, 0, stream,
                       ts, control, dWn, Wh, Wu, bh, sigma, Wo, bo, out);
}